// Sampler_16784732193183
// MI455X (gfx1250) — compile-verified
//
#include <hip/hip_runtime.h>
#include <stdint.h>

// Sampler: B=256 rows, V=128256 vocab.
// out[row] = greedy ? argmax(logits) : argmax(logits/t - log(max(-log u, 1e-10)))
// Single streaming pass, async-to-LDS pipelined (CDNA5 ASYNCcnt path).

#define VOCAB   128256
#define BLK     256
#define WAVES   (BLK / 32)
#define STAGES  6
#define CHUNKS  (VOCAB / 4)              // 32064 float4 chunks per row
#define ITER_LO (CHUNKS / BLK)           // 125
#define REM     (CHUNKS - ITER_LO * BLK) // 64 -> threads 0..63 do one extra

// Map float -> uint such that larger float => larger uint (totally ordered).
__device__ __forceinline__ uint32_t orderedFloat(float f) {
    uint32_t b = __float_as_uint(f);
    return (b & 0x80000000u) ? ~b : (b | 0x80000000u);
}

// Pack (value, index): u64 max picks max value; on ties picks smallest index.
__device__ __forceinline__ unsigned long long packVI(float v, uint32_t idx) {
    return ((unsigned long long)orderedFloat(v) << 32) |
           (unsigned long long)(0xFFFFFFFFu - idx);
}

__device__ __forceinline__ uint32_t unpackIdx(unsigned long long p) {
    return 0xFFFFFFFFu - (uint32_t)(p & 0xFFFFFFFFull);
}

__global__ __launch_bounds__(BLK) void Sampler_16784732193183_kernel(
    const float* __restrict__ logits,
    const float* __restrict__ temps,
    const float* __restrict__ u,
    int* __restrict__ out)
{
    __shared__ float4 sL[STAGES][BLK];   // logits staging (24 KB)
    __shared__ float4 sU[STAGES][BLK];   // uniform-draw staging (24 KB)
    __shared__ unsigned long long redG[WAVES];
    __shared__ unsigned long long redS[WAVES];

    const int tid = threadIdx.x;
    const int row = blockIdx.x;

    const float* lrow = logits + (size_t)row * VOCAB;
    const float* urow = u      + (size_t)row * VOCAB;

    const float tv     = temps[row];
    const bool  greedy = (tv <= 1e-10f);
    const float invT   = greedy ? 1.0f : (1.0f / tv);

    const int nIter = (tid < REM) ? (ITER_LO + 1) : ITER_LO; // wave-uniform

    // LDS byte addresses for this lane's slots (low 32 bits of flat shared ptr
    // == addrspace(3) offset; hardware adds LDS_BASE).
    uint32_t ldsL[STAGES], ldsU[STAGES];
#pragma unroll
    for (int s = 0; s < STAGES; ++s) {
        ldsL[s] = (uint32_t)(uintptr_t)&sL[s][tid];
        ldsU[s] = (uint32_t)(uintptr_t)&sU[s][tid];
    }

    // Issue one 16B async load per stream into a stage buffer (GVS addressing:
    // SGPR base + per-lane 32-bit byte offset). Tracked by ASYNCcnt.
#define ISSUE_STAGE(iter, stage)                                                \
    do {                                                                        \
        uint32_t goff = ((uint32_t)tid + (uint32_t)(iter) * BLK) * 16u;         \
        asm volatile("global_load_async_to_lds_b128 %0, %1, %2"                 \
                     :: "v"(ldsL[(stage)]), "v"(goff), "s"(lrow) : "memory");   \
        asm volatile("global_load_async_to_lds_b128 %0, %1, %2"                 \
                     :: "v"(ldsU[(stage)]), "v"(goff), "s"(urow) : "memory");   \
    } while (0)

    unsigned long long bestG = 0ull; // packed (-inf) sentinel: any real beats 0
    unsigned long long bestS = 0ull;

    // Consume one staged chunk (assumes its async loads have completed).
#define CONSUME(I, STAGE)                                                       \
    do {                                                                        \
        float4 l_  = sL[(STAGE)][tid];                                          \
        float4 uv_ = sU[(STAGE)][tid];                                          \
        const uint32_t vbase_ = ((uint32_t)tid + (uint32_t)(I) * BLK) * 4u;     \
        PROCESS(l_.x, uv_.x, vbase_ + 0u);                                      \
        PROCESS(l_.y, uv_.y, vbase_ + 1u);                                      \
        PROCESS(l_.z, uv_.z, vbase_ + 2u);                                      \
        PROCESS(l_.w, uv_.w, vbase_ + 3u);                                      \
    } while (0)

#define PROCESS(LV, UV, VI)                                                     \
    do {                                                                        \
        float lv_ = (LV);                                                       \
        unsigned long long pg_ = packVI(lv_, (VI));                             \
        if (pg_ > bestG) bestG = pg_;                                           \
        float noise_ = fmaxf(-__logf((UV)), 1e-10f);                            \
        float key_   = lv_ * invT - __logf(noise_);                             \
        unsigned long long ps_ = packVI(key_, (VI));                            \
        if (ps_ > bestS) bestS = ps_;                                           \
    } while (0)

    // ---- Prime the pipeline (all threads have >= STAGES iterations). ----
#pragma unroll
    for (int s = 0; s < STAGES; ++s) ISSUE_STAGE(s, s);

    // ---- Main loop: steady-state wait is valid only while >= STAGES-1
    //      younger stages remain in flight, i.e. for i <= nIter-STAGES. ----
    const int nMain = nIter - (STAGES - 1);
    int i = 0;
    for (; i < nMain; ++i) {
        const int stage = i % STAGES;

        // Oldest stage's 2 loads complete once <= 2*(STAGES-1) remain in
        // flight (per-wave async loads complete in order).
        asm volatile("s_wait_asynccnt %0" :: "i"(2 * (STAGES - 1)) : "memory");

        CONSUME(i, stage);

        // Ensure our ds_loads finished before the async engine may overwrite
        // this stage's buffer with the next issue.
        asm volatile("s_wait_dscnt 0" ::: "memory");

        const int nx = i + STAGES;
        if (nx < nIter) ISSUE_STAGE(nx, stage);
    }

    // ---- Epilogue: fewer than STAGES-1 younger stages in flight, so the
    //      steady-state immediate would pass early. Drain fully instead
    //      (first iteration pays; the rest see ASYNCcnt == 0). ----
    for (; i < nIter; ++i) {
        const int stage = i % STAGES;
        asm volatile("s_wait_asynccnt 0" ::: "memory");
        CONSUME(i, stage);
    }

#undef PROCESS
#undef CONSUME
#undef ISSUE_STAGE

    // Wave32 butterfly reduction (u64 max == max-value / min-index).
#pragma unroll
    for (int m = 16; m >= 1; m >>= 1) {
        unsigned long long og = __shfl_xor(bestG, m, 32);
        unsigned long long os = __shfl_xor(bestS, m, 32);
        if (og > bestG) bestG = og;
        if (os > bestS) bestS = os;
    }

    const int wave = tid >> 5;
    if ((tid & 31) == 0) { redG[wave] = bestG; redS[wave] = bestS; }
    __syncthreads();

    if (tid == 0) {
        unsigned long long g = redG[0], s = redS[0];
#pragma unroll
        for (int w = 1; w < WAVES; ++w) {
            if (redG[w] > g) g = redG[w];
            if (redS[w] > s) s = redS[w];
        }
        out[row] = (int)(greedy ? unpackIdx(g) : unpackIdx(s));
    }
}

extern "C" void kernel_launch(void* const* d_in, const int* in_sizes, int n_in,
                              void* d_out, int out_size, void* d_ws, size_t ws_size,
                              hipStream_t stream) {
    const float* logits = (const float*)d_in[0];
    const float* temps  = (const float*)d_in[1];
    const float* u      = (const float*)d_in[2];
    int* out = (int*)d_out;
    const int B = in_sizes[1]; // 256 rows (one workgroup per row)
    Sampler_16784732193183_kernel<<<B, BLK, 0, stream>>>(logits, temps, u, out);
}